// AttentivePooling_11304353923437
// MI455X (gfx1250) — compile-verified
//
#include <hip/hip_runtime.h>
#include <math.h>

#define DIM 256          // feature dim == out dim
#define WAVES_PER_BLOCK 8
#define NT 8             // column tiles (of 16) per wave -> 128 cols per wave

typedef __attribute__((ext_vector_type(2))) float v2f;
typedef __attribute__((ext_vector_type(8))) float v8f;

__device__ __forceinline__ float fast_tanh(float x) {
    // overflow-safe tanh via exp of negative magnitude + single v_rcp_f32
    float e = __expf(-2.0f * fabsf(x));
    float t = (1.0f - e) * __builtin_amdgcn_rcpf(1.0f + e);
    return copysignf(t, x);
}

// -----------------------------------------------------------------------------
// Kernel 1: h = tanh(X @ W1 + b1) using V_WMMA_F32_16X16X4_F32.
// One wave per 16x128 output stripe: A fragment loaded once per K-step and
// reused across NT=8 WMMAs into independent accumulators (no D->A/B hazard).
// A (16x4 f32): lane-half hf, VGPR v holds K = v + 2*hf of row M = lane&15.
// B (4x16 f32): lane-half hf, VGPR v holds row K = v + 2*hf, col N = lane&15.
// C/D (16x16 f32): lane-half hf, VGPR v holds M = v + 8*hf, N = lane&15.
// -----------------------------------------------------------------------------
__global__ void __launch_bounds__(256)
wmma_h_kernel(const float* __restrict__ X, const float* __restrict__ W1,
              const float* __restrict__ b1, float* __restrict__ h, int nrows)
{
    const int lane = threadIdx.x & 31;
    const int wave = threadIdx.x >> 5;
    const int tile = blockIdx.x * WAVES_PER_BLOCK + wave;
    const int groups = DIM / (16 * NT);              // 2 column groups of 128
    const int mtile = tile / groups;
    const int g     = tile % groups;
    if (mtile * 16 >= nrows) return;

    const int half = lane >> 4;   // 0 or 1
    const int l16  = lane & 15;

    const float* arow  = X  + (size_t)(mtile * 16 + l16) * DIM;
    const float* bbase = W1 + g * (16 * NT) + l16;

    v8f acc[NT];
#pragma unroll
    for (int t = 0; t < NT; ++t) acc[t] = (v8f){};

#pragma unroll 2
    for (int k = 0; k < DIM; k += 4) {
        const int ka = k + 2 * half;
        v2f afrag;
        afrag.x = arow[ka];
        afrag.y = arow[ka + 1];
#pragma unroll
        for (int t = 0; t < NT; ++t) {
            v2f bfrag;
            bfrag.x = bbase[(size_t)ka * DIM + t * 16];
            bfrag.y = bbase[(size_t)(ka + 1) * DIM + t * 16];
            acc[t] = __builtin_amdgcn_wmma_f32_16x16x4_f32(
                /*neg_a=*/false, afrag, /*neg_b=*/false, bfrag,
                /*c_mod=*/(short)0, acc[t], /*reuse_a=*/false, /*reuse_b=*/false);
        }
    }

#pragma unroll
    for (int t = 0; t < NT; ++t) {
        const int col  = g * (16 * NT) + t * 16 + l16;
        const float bias = b1[col];
#pragma unroll
        for (int v = 0; v < 8; ++v) {
            const int r = mtile * 16 + v + 8 * half;   // full tiles: no guard
            h[(size_t)r * DIM + col] = fast_tanh(acc[t][v] + bias);
        }
    }
}

// -----------------------------------------------------------------------------
// Kernel 2: a[n] = X[n,:] . W2 + b2  (one block of 256 threads per row)
// -----------------------------------------------------------------------------
__global__ void __launch_bounds__(256)
attn_score_kernel(const float* __restrict__ X, const float* __restrict__ W2,
                  const float* __restrict__ b2, float* __restrict__ a, int nrows)
{
    __shared__ float red[DIM];
    const int row = blockIdx.x;
    const int tid = threadIdx.x;
    if (row >= nrows) return;
    red[tid] = X[(size_t)row * DIM + tid] * W2[tid];
    __syncthreads();
#pragma unroll
    for (int off = DIM / 2; off > 0; off >>= 1) {
        if (tid < off) red[tid] += red[tid + off];
        __syncthreads();
    }
    if (tid == 0) a[row] = red[0] + b2[0];
}

// -----------------------------------------------------------------------------
// Kernel 3: per-molecule softmax over scores s = mnm*a + mask, write w,
// and pooled[m] = sum_n w[m,n] * h[n,:] (skipping exact-zero weights).
// One block (256 threads) per molecule; thread t owns output dim t.
// -----------------------------------------------------------------------------
__global__ void __launch_bounds__(256)
softmax_pool_kernel(const float* __restrict__ mnm, const float* __restrict__ mask,
                    const float* __restrict__ a, const float* __restrict__ h,
                    float* __restrict__ wout, float* __restrict__ pooled, int N)
{
    const int m   = blockIdx.x;
    const int tid = threadIdx.x;
    const float* rowm = mnm  + (size_t)m * N;
    const float* rowk = mask + (size_t)m * N;

    __shared__ float redm[256];
    __shared__ float reds[256];
    __shared__ int   cnt;
    __shared__ int   sidx[256];
    __shared__ float sw[256];

    // ---- pass 1: online (max, sum-of-exp) over this molecule's row ----
    float mloc = -INFINITY, sloc = 0.0f;
    for (int n = tid; n < N; n += 256) {
        __builtin_prefetch(&rowm[n + 2048], 0, 1);
        __builtin_prefetch(&rowk[n + 2048], 0, 1);
        float sc   = fmaf(rowm[n], a[n], rowk[n]);
        float mnew = fmaxf(mloc, sc);
        sloc = sloc * __expf(mloc - mnew) + __expf(sc - mnew);
        mloc = mnew;
    }
    redm[tid] = mloc;
    reds[tid] = sloc;
    __syncthreads();
    for (int off = 128; off > 0; off >>= 1) {
        if (tid < off) {
            float ma = redm[tid], mb = redm[tid + off];
            float sa = reds[tid], sb = reds[tid + off];
            float mm = fmaxf(ma, mb);
            redm[tid] = mm;
            reds[tid] = sa * __expf(ma - mm) + sb * __expf(mb - mm);
        }
        __syncthreads();
    }
    const float rowmax = redm[0];
    const float invsum = 1.0f / reds[0];
    __syncthreads();

    // ---- pass 2: write w; compact nonzero weights; accumulate pooled ----
    float acc = 0.0f;   // thread tid owns output dim tid
    for (int n0 = 0; n0 < N; n0 += 256) {
        if (tid == 0) cnt = 0;
        __syncthreads();
        const int n = n0 + tid;
        if (n < N) {
            __builtin_prefetch(&rowm[n + 2048], 0, 1);
            __builtin_prefetch(&rowk[n + 2048], 0, 1);
            float sc = fmaf(rowm[n], a[n], rowk[n]);
            float wv = __expf(sc - rowmax) * invsum;
            wout[(size_t)m * N + n] = wv;
            if (wv != 0.0f) {             // exp underflow -> exactly 0 for masked
                int p = atomicAdd(&cnt, 1);
                sidx[p] = n;
                sw[p]   = wv;
            }
        }
        __syncthreads();
        const int c = cnt;
        for (int j = 0; j < c; ++j)
            acc = fmaf(sw[j], h[(size_t)sidx[j] * DIM + tid], acc);
        __syncthreads();
    }
    pooled[(size_t)m * DIM + tid] = acc;
}

// -----------------------------------------------------------------------------
// Host launcher
// -----------------------------------------------------------------------------
extern "C" void kernel_launch(void* const* d_in, const int* in_sizes, int n_in,
                              void* d_out, int out_size, void* d_ws, size_t ws_size,
                              hipStream_t stream) {
    const float* X    = (const float*)d_in[0];   // [N, 256]
    const float* mnm  = (const float*)d_in[1];   // [M, N]
    const float* mask = (const float*)d_in[2];   // [M, N]
    const float* W1   = (const float*)d_in[3];   // [256, 256]
    const float* b1   = (const float*)d_in[4];   // [256]
    const float* W2   = (const float*)d_in[5];   // [256]
    const float* b2   = (const float*)d_in[6];   // [1]

    const int N = in_sizes[0] / DIM;             // 50000 nodes
    const int M = in_sizes[1] / N;               // 1000 molecules

    float* h = (float*)d_ws;                     // [N, 256] workspace
    float* a = h + (size_t)N * DIM;              // [N] workspace

    float* pooled = (float*)d_out;               // [M, 256]
    float* wout   = pooled + (size_t)M * DIM;    // [M, N]

    // h = tanh(X @ W1 + b1) via WMMA (one wave per 16x128 stripe)
    const int mtiles  = (N + 15) / 16;
    const int tiles   = mtiles * (DIM / (16 * NT));
    const int blocksA = (tiles + WAVES_PER_BLOCK - 1) / WAVES_PER_BLOCK;
    wmma_h_kernel<<<blocksA, 256, 0, stream>>>(X, W1, b1, h, N);

    // a = X @ W2 + b2
    attn_score_kernel<<<N, 256, 0, stream>>>(X, W2, b2, a, N);

    // softmax + w writeout + pooled
    softmax_pool_kernel<<<M, 256, 0, stream>>>(mnm, mask, a, h, wout, pooled, N);
}